// MyShader_81793357185200
// MI455X (gfx1250) — compile-verified
//
#include <hip/hip_runtime.h>
#include <cstdint>
#include <cstddef>

// Pure gather / blend workload: 0 matrix FLOPs, ~0.68 GB of HBM traffic
// => bandwidth-bound (~29us floor at 23.3 TB/s). Optimization levers on
// MI455X: b128 vector memory ops, non-temporal cache policy for the 416MB
// of streaming stores (protect L2 residency of the 6.4MB gather tables),
// and global_prefetch_b8 on the streaming inputs.

typedef __attribute__((ext_vector_type(4))) float v4f;
typedef __attribute__((ext_vector_type(4))) int   v4i;

#define KF 8  // faces per pixel (K in the reference)

// ---------------------------------------------------------------------------
// Kernel A: one thread per face. Compute centroid once (vs ~40x redundant
// per-pixel recompute) and repack centroids+normals as float4 so the hot
// per-pixel gathers are single aligned global_load_b128 from L2.
// ---------------------------------------------------------------------------
__global__ __launch_bounds__(256)
void face_tables_kernel(const float* __restrict__ verts,
                        const int*   __restrict__ faces,
                        const float* __restrict__ fnorm,
                        v4f* __restrict__ cent_tab,
                        v4f* __restrict__ norm_tab,
                        int F)
{
    int f = blockIdx.x * 256 + threadIdx.x;
    if (f >= F) return;

    int i0 = faces[3 * f + 0];
    int i1 = faces[3 * f + 1];
    int i2 = faces[3 * f + 2];
    const float s = 1.0f / 3.0f;

    v4f c;
    c.x = (verts[3 * i0 + 0] + verts[3 * i1 + 0] + verts[3 * i2 + 0]) * s;
    c.y = (verts[3 * i0 + 1] + verts[3 * i1 + 1] + verts[3 * i2 + 1]) * s;
    c.z = (verts[3 * i0 + 2] + verts[3 * i1 + 2] + verts[3 * i2 + 2]) * s;
    c.w = 0.0f;
    cent_tab[f] = c;

    v4f n;
    n.x = fnorm[3 * f + 0];
    n.y = fnorm[3 * f + 1];
    n.z = fnorm[3 * f + 2];
    n.w = 0.0f;
    norm_tab[f] = n;
}

// ---------------------------------------------------------------------------
// Kernel B: one thread per pixel (N*H*W). Handles all K=8 slots so the
// 96-byte coords/normals outputs per pixel become 6 contiguous b128 NT
// stores each; folds in the hard_rgb_blend (images) output.
// USE_WS=false fallback recomputes centroids inline if scratch is too small.
// ---------------------------------------------------------------------------
template <bool USE_WS>
__global__ __launch_bounds__(256)
void shade_kernel(const int*   __restrict__ pix,
                  const float* __restrict__ texels,
                  const v4f*   __restrict__ cent_tab,
                  const v4f*   __restrict__ norm_tab,
                  const float* __restrict__ verts,
                  const int*   __restrict__ faces,
                  const float* __restrict__ fnorm,
                  float* __restrict__ images,
                  float* __restrict__ coords,
                  float* __restrict__ pnorms,
                  int NHW)
{
    int p = blockIdx.x * 256 + threadIdx.x;
    if (p >= NHW) return;

    const int* pslot = pix + (size_t)p * KF;

    // Stream-ahead prefetch of the two big sequential inputs (global_prefetch_b8).
    // Speculative: out-of-range translations are silently dropped by HW.
    __builtin_prefetch(pslot + 2048 * KF, 0, 1);
    __builtin_prefetch(texels + (size_t)(p + 2048) * (KF * 3), 0, 1);

    // 8 face indices: two aligned int4 loads, non-temporal (single use).
    v4i idx_lo = __builtin_nontemporal_load((const v4i*)pslot);
    v4i idx_hi = __builtin_nontemporal_load((const v4i*)pslot + 1);
    int idx[KF] = { idx_lo.x, idx_lo.y, idx_lo.z, idx_lo.w,
                    idx_hi.x, idx_hi.y, idx_hi.z, idx_hi.w };

    float oc[KF * 3];
    float on[KF * 3];

#pragma unroll
    for (int k = 0; k < KF; ++k) {
        int f = idx[k];
        float cx = 0.f, cy = 0.f, cz = 0.f;
        float nx = 0.f, ny = 0.f, nz = 0.f;
        if (f >= 0) {
            if (USE_WS) {
                v4f c = cent_tab[f];   // b128 gather, L2-resident table
                v4f n = norm_tab[f];
                cx = c.x; cy = c.y; cz = c.z;
                nx = n.x; ny = n.y; nz = n.z;
            } else {
                int i0 = faces[3 * f + 0];
                int i1 = faces[3 * f + 1];
                int i2 = faces[3 * f + 2];
                const float s = 1.0f / 3.0f;
                cx = (verts[3 * i0 + 0] + verts[3 * i1 + 0] + verts[3 * i2 + 0]) * s;
                cy = (verts[3 * i0 + 1] + verts[3 * i1 + 1] + verts[3 * i2 + 1]) * s;
                cz = (verts[3 * i0 + 2] + verts[3 * i1 + 2] + verts[3 * i2 + 2]) * s;
                nx = fnorm[3 * f + 0];
                ny = fnorm[3 * f + 1];
                nz = fnorm[3 * f + 2];
            }
        }
        oc[3 * k + 0] = cx; oc[3 * k + 1] = cy; oc[3 * k + 2] = cz;
        on[3 * k + 0] = nx; on[3 * k + 1] = ny; on[3 * k + 2] = nz;
    }

    // 96B contiguous per output per pixel -> 6 NT b128 stores each.
    float* cdst = coords + (size_t)p * (KF * 3);
    float* ndst = pnorms + (size_t)p * (KF * 3);
#pragma unroll
    for (int q = 0; q < 6; ++q) {
        v4f vc = { oc[4 * q + 0], oc[4 * q + 1], oc[4 * q + 2], oc[4 * q + 3] };
        v4f vn = { on[4 * q + 0], on[4 * q + 1], on[4 * q + 2], on[4 * q + 3] };
        __builtin_nontemporal_store(vc, (v4f*)(cdst + 4 * q));
        __builtin_nontemporal_store(vn, (v4f*)(ndst + 4 * q));
    }

    // hard_rgb_blend: texels[..., 0, :3] vs background (1,1,1), alpha = !bg.
    // Aligned float4 load (4th lane is texels[...,1,0], always in-bounds).
    v4f t = __builtin_nontemporal_load((const v4f*)(texels + (size_t)p * (KF * 3)));
    bool bg = idx[0] < 0;
    v4f img;
    img.x = bg ? 1.0f : t.x;
    img.y = bg ? 1.0f : t.y;
    img.z = bg ? 1.0f : t.z;
    img.w = bg ? 0.0f : 1.0f;
    __builtin_nontemporal_store(img, (v4f*)(images + (size_t)p * 4));
}

// ---------------------------------------------------------------------------
extern "C" void kernel_launch(void* const* d_in, const int* in_sizes, int n_in,
                              void* d_out, int out_size, void* d_ws, size_t ws_size,
                              hipStream_t stream)
{
    const float* verts  = (const float*)d_in[0];   // (V,3) f32
    const int*   faces  = (const int*)  d_in[1];   // (F,3) i32
    const float* fnorm  = (const float*)d_in[2];   // (F,3) f32
    const int*   pix    = (const int*)  d_in[3];   // (N,H,W,K) i32
    const float* texels = (const float*)d_in[4];   // (N,H,W,K,3) f32

    const int F     = in_sizes[1] / 3;
    const int nslot = in_sizes[3];        // N*H*W*K
    const int NHW   = nslot / KF;         // one thread per pixel

    // Output tuple, flat in return order:
    //   images (N,H,W,4) | pixel_coords (N,H,W,K,3) | pixel_normals (N,H,W,K,3)
    float* images = (float*)d_out;
    float* coords = images + (size_t)NHW * 4;
    float* pnorms = coords + (size_t)nslot * 3;

    v4f* cent_tab = (v4f*)d_ws;
    v4f* norm_tab = cent_tab + F;
    const size_t ws_need = (size_t)2 * (size_t)F * sizeof(v4f);  // 6.4 MB
    const bool use_ws = (d_ws != nullptr) && (ws_size >= ws_need);

    const int blk = 256;
    if (use_ws) {
        face_tables_kernel<<<(F + blk - 1) / blk, blk, 0, stream>>>(
            verts, faces, fnorm, cent_tab, norm_tab, F);
        shade_kernel<true><<<(NHW + blk - 1) / blk, blk, 0, stream>>>(
            pix, texels, cent_tab, norm_tab, verts, faces, fnorm,
            images, coords, pnorms, NHW);
    } else {
        shade_kernel<false><<<(NHW + blk - 1) / blk, blk, 0, stream>>>(
            pix, texels, nullptr, nullptr, verts, faces, fnorm,
            images, coords, pnorms, NHW);
    }
}